// MultiLevelHybridHashEncoding_13907104104961
// MI455X (gfx1250) — compile-verified
//
#include <hip/hip_runtime.h>
#include <stdint.h>

// ---------------------------------------------------------------------------
// Instant-NGP multilevel hash encoding for MI455X (gfx1250, wave32).
//   B = 1,000,000 points, 16 levels, 2 features/entry.
//   Levels 0-7 dense (res^3 <= 2^19), levels 8-15 hashed (N_ENC = 2^19).
//   Levels 0-1 tables (32KB + 64KB) staged to LDS via the Tensor Data Mover.
// ---------------------------------------------------------------------------

#define NPOINTS   1000000
#define NBLOCKS   1024
#define NTHREADS  256

typedef float    f32x2  __attribute__((ext_vector_type(2)));
typedef uint32_t u32x4v __attribute__((ext_vector_type(4)));
typedef uint32_t u32x8v __attribute__((ext_vector_type(8)));

// ---- Tensor Data Mover: 1-row 2D tile, data_size = 4B, contiguous copy ----
// D# layout per CDNA5 ISA ch.10.8 (group0 = 4 SGPRs, group1 = 8 SGPRs).
// The LDS destination pointer is passed as an (unused) VGPR operand so the
// shared array ESCAPES into the asm: without it, LLVM proves the non-escaping
// shared array is never stored and folds all ds_loads from it to poison.
__device__ __forceinline__ void tdm_load_to_lds(float* lds_ptr,
                                                uint32_t lds_byte_off,
                                                const void* gsrc,
                                                uint32_t nfloats) {
  const uint64_t ga = (uint64_t)(uintptr_t)gsrc;
  u32x4v g0;
  g0[0] = 1u;                                              // count=1, user D#
  g0[1] = lds_byte_off;                                    // lds_addr
  g0[2] = (uint32_t)ga;                                    // global_addr[31:0]
  g0[3] = ((uint32_t)(ga >> 32) & 0x1FFFFFFu) | (2u << 30); // addr[56:32] | type=2
  u32x8v g1;
  g1[0] = (2u << 16);                        // wg_mask=0, data_size=2 (4 bytes)
  g1[1] = (nfloats & 0xFFFFu) << 16;         // tensor_dim0[15:0]
  g1[2] = (nfloats >> 16) | (1u << 16);      // tensor_dim0[31:16] | tensor_dim1=1
  g1[3] = (nfloats & 0xFFFFu) << 16;         // tensor_dim1 hi=0 | tile_dim0
  g1[4] = 1u;                                // tile_dim1=1, tile_dim2=0
  g1[5] = nfloats;                           // tensor_dim0_stride[31:0]
  g1[6] = (nfloats & 0xFFFFu) << 16;         // stride0 hi=0 | tensor_dim1_stride lo
  g1[7] = (nfloats >> 16);                   // tensor_dim1_stride hi
  asm volatile("tensor_load_to_lds %0, %1"
               :: "s"(g0), "s"(g1), "v"(lds_ptr)
               : "memory");
}

__device__ __forceinline__ void wait_tensorcnt0() {
#if __has_builtin(__builtin_amdgcn_s_wait_tensorcnt)
  __builtin_amdgcn_s_wait_tensorcnt(0);
#else
  asm volatile("s_wait_tensorcnt 0x0" ::: "memory");
#endif
}

// ---- one level: trilinear gather-interpolate --------------------------------
// Masking trick: out-of-range per-dim corners get weight 0 (exactly matching
// the reference's where(m, w, 0)); indices are clamped (dense) or naturally
// in-range (hash: & (2^19-1)), so no OOB access and gathered value is ignored.
template<int RES, bool HASH>
__device__ __forceinline__ f32x2 level_feat(const float* __restrict__ tab,
                                            float px, float py, float pz) {
  constexpr float S = (float)RES * 0.5f;
  const float xp = fmaf(px, S, S - 0.5f);
  const float yp = fmaf(py, S, S - 0.5f);
  const float zp = fmaf(pz, S, S - 0.5f);
  const float fx = floorf(xp), fy = floorf(yp), fz = floorf(zp);
  const float tx = xp - fx, ty = yp - fy, tz = zp - fz;
  const int ix = (int)fx, iy = (int)fy, iz = (int)fz;
  const int jx = ix + 1, jy = iy + 1, jz = iz + 1;

  const float wx0 = ((unsigned)ix < (unsigned)RES) ? 1.0f - tx : 0.0f;
  const float wx1 = ((unsigned)jx < (unsigned)RES) ? tx        : 0.0f;
  const float wy0 = ((unsigned)iy < (unsigned)RES) ? 1.0f - ty : 0.0f;
  const float wy1 = ((unsigned)jy < (unsigned)RES) ? ty        : 0.0f;
  const float wz0 = ((unsigned)iz < (unsigned)RES) ? 1.0f - tz : 0.0f;
  const float wz1 = ((unsigned)jz < (unsigned)RES) ? tz        : 0.0f;

  uint32_t i000, i100, i010, i110, i001, i101, i011, i111;
  if constexpr (HASH) {
    const uint32_t M   = 0x7FFFFu;                 // N_ENC = 2^19
    const uint32_t hx0 = (uint32_t)ix,                hx1 = (uint32_t)jx;
    const uint32_t hy0 = (uint32_t)iy * 2654435761u, hy1 = (uint32_t)jy * 2654435761u;
    const uint32_t hz0 = (uint32_t)iz * 805459861u,  hz1 = (uint32_t)jz * 805459861u;
    i000 = (hx0 ^ hy0 ^ hz0) & M;  i100 = (hx1 ^ hy0 ^ hz0) & M;
    i010 = (hx0 ^ hy1 ^ hz0) & M;  i110 = (hx1 ^ hy1 ^ hz0) & M;
    i001 = (hx0 ^ hy0 ^ hz1) & M;  i101 = (hx1 ^ hy0 ^ hz1) & M;
    i011 = (hx0 ^ hy1 ^ hz1) & M;  i111 = (hx1 ^ hy1 ^ hz1) & M;
  } else {
    const int cx0 = ix < 0 ? 0 : ix;
    const int cx1 = jx > RES - 1 ? RES - 1 : jx;
    const int cy0 = iy < 0 ? 0 : iy;
    const int cy1 = jy > RES - 1 ? RES - 1 : jy;
    const int cz0 = iz < 0 ? 0 : iz;
    const int cz1 = jz > RES - 1 ? RES - 1 : jz;
    const int ry0 = cy0 * RES,          ry1 = cy1 * RES;
    const int rz0 = cz0 * (RES * RES),  rz1 = cz1 * (RES * RES);
    i000 = (uint32_t)(cx0 + ry0 + rz0);  i100 = (uint32_t)(cx1 + ry0 + rz0);
    i010 = (uint32_t)(cx0 + ry1 + rz0);  i110 = (uint32_t)(cx1 + ry1 + rz0);
    i001 = (uint32_t)(cx0 + ry0 + rz1);  i101 = (uint32_t)(cx1 + ry0 + rz1);
    i011 = (uint32_t)(cx0 + ry1 + rz1);  i111 = (uint32_t)(cx1 + ry1 + rz1);
  }

  const f32x2* __restrict__ t2 = (const f32x2*)tab;
  const f32x2 e000 = t2[i000], e100 = t2[i100];
  const f32x2 e010 = t2[i010], e110 = t2[i110];
  const f32x2 e001 = t2[i001], e101 = t2[i101];
  const f32x2 e011 = t2[i011], e111 = t2[i111];

  const float wz0y0 = wz0 * wy0, wz0y1 = wz0 * wy1;
  const float wz1y0 = wz1 * wy0, wz1y1 = wz1 * wy1;
  const float w000 = wx0 * wz0y0, w100 = wx1 * wz0y0;
  const float w010 = wx0 * wz0y1, w110 = wx1 * wz0y1;
  const float w001 = wx0 * wz1y0, w101 = wx1 * wz1y0;
  const float w011 = wx0 * wz1y1, w111 = wx1 * wz1y1;

  // two independent FMA chains to cut dependent latency
  float a0 = w000 * e000[0], a1 = w000 * e000[1];
  float b0 = w100 * e100[0], b1 = w100 * e100[1];
  a0 = fmaf(w010, e010[0], a0);  a1 = fmaf(w010, e010[1], a1);
  b0 = fmaf(w110, e110[0], b0);  b1 = fmaf(w110, e110[1], b1);
  a0 = fmaf(w001, e001[0], a0);  a1 = fmaf(w001, e001[1], a1);
  b0 = fmaf(w101, e101[0], b0);  b1 = fmaf(w101, e101[1], b1);
  a0 = fmaf(w011, e011[0], a0);  a1 = fmaf(w011, e011[1], a1);
  b0 = fmaf(w111, e111[0], b0);  b1 = fmaf(w111, e111[1], b1);
  f32x2 r; r[0] = a0 + b0; r[1] = a1 + b1;
  return r;
}

__global__ __launch_bounds__(NTHREADS)
void hashenc_kernel(const float* __restrict__ x,
                    const float* __restrict__ e0,  const float* __restrict__ e1,
                    const float* __restrict__ e2,  const float* __restrict__ e3,
                    const float* __restrict__ e4,  const float* __restrict__ e5,
                    const float* __restrict__ e6,  const float* __restrict__ e7,
                    const float* __restrict__ e8,  const float* __restrict__ e9,
                    const float* __restrict__ e10, const float* __restrict__ e11,
                    const float* __restrict__ e12, const float* __restrict__ e13,
                    const float* __restrict__ e14, const float* __restrict__ e15,
                    float* __restrict__ out) {
  // level0: 16^3*2 = 8192 floats (32KB); level1: 20^3*2 = 16000 floats (64KB)
  __shared__ __align__(16) float s_emb[8192 + 16000];

  if (threadIdx.x < 32u) {          // wave 0 of the workgroup issues the DMA
    tdm_load_to_lds(s_emb,        0u,     e0, 8192u);
    tdm_load_to_lds(s_emb + 8192, 32768u, e1, 16000u);
    wait_tensorcnt0();
  }
  __syncthreads();

  const int tid    = (int)(blockIdx.x * NTHREADS + threadIdx.x);
  const int stride = NBLOCKS * NTHREADS;

#pragma clang loop unroll(disable)
  for (int b = tid; b < NPOINTS; b += stride) {
    const int bn = b + stride;
    if (bn < NPOINTS) __builtin_prefetch(x + 3 * (size_t)bn, 0, 0);

    const float px = x[3 * (size_t)b + 0];
    const float py = x[3 * (size_t)b + 1];
    const float pz = x[3 * (size_t)b + 2];
    float* __restrict__ o = out + (size_t)b * 32;

    // LDS-resident levels (ds_load_b64 gathers)
    *(f32x2*)(o + 0)  = level_feat<16,  false>(s_emb,        px, py, pz);
    *(f32x2*)(o + 2)  = level_feat<20,  false>(s_emb + 8192, px, py, pz);
    // dense levels from L2
    *(f32x2*)(o + 4)  = level_feat<25,  false>(e2,  px, py, pz);
    *(f32x2*)(o + 6)  = level_feat<32,  false>(e3,  px, py, pz);
    *(f32x2*)(o + 8)  = level_feat<40,  false>(e4,  px, py, pz);
    *(f32x2*)(o + 10) = level_feat<50,  false>(e5,  px, py, pz);
    *(f32x2*)(o + 12) = level_feat<64,  false>(e6,  px, py, pz);
    *(f32x2*)(o + 14) = level_feat<80,  false>(e7,  px, py, pz);
    // hashed levels (id = (x ^ y*P1 ^ z*P2) & (2^19-1))
    *(f32x2*)(o + 16) = level_feat<101, true>(e8,  px, py, pz);
    *(f32x2*)(o + 18) = level_feat<128, true>(e9,  px, py, pz);
    *(f32x2*)(o + 20) = level_feat<161, true>(e10, px, py, pz);
    *(f32x2*)(o + 22) = level_feat<203, true>(e11, px, py, pz);
    *(f32x2*)(o + 24) = level_feat<256, true>(e12, px, py, pz);
    *(f32x2*)(o + 26) = level_feat<322, true>(e13, px, py, pz);
    *(f32x2*)(o + 28) = level_feat<406, true>(e14, px, py, pz);
    *(f32x2*)(o + 30) = level_feat<512, true>(e15, px, py, pz);
  }
}

extern "C" void kernel_launch(void* const* d_in, const int* in_sizes, int n_in,
                              void* d_out, int out_size, void* d_ws, size_t ws_size,
                              hipStream_t stream) {
  (void)in_sizes; (void)n_in; (void)out_size; (void)d_ws; (void)ws_size;
  const float* x = (const float*)d_in[0];
  hashenc_kernel<<<NBLOCKS, NTHREADS, 0, stream>>>(
      x,
      (const float*)d_in[1],  (const float*)d_in[2],
      (const float*)d_in[3],  (const float*)d_in[4],
      (const float*)d_in[5],  (const float*)d_in[6],
      (const float*)d_in[7],  (const float*)d_in[8],
      (const float*)d_in[9],  (const float*)d_in[10],
      (const float*)d_in[11], (const float*)d_in[12],
      (const float*)d_in[13], (const float*)d_in[14],
      (const float*)d_in[15], (const float*)d_in[16],
      (float*)d_out);
}